// DecoderLSTM_86053964742824
// MI455X (gfx1250) — compile-verified
//
#include <hip/hip_runtime.h>
#include <hip/hip_bf16.h>
#include <math.h>

#define HDIM 1024
#define VDIM 50257
#define GDIM (4 * HDIM)          // 4096 gate rows
#define NTILES_OUT ((VDIM + 15) / 16)   // 3142
#define LSE_BLOCKS 64
#define LSE_CHUNK ((VDIM + LSE_BLOCKS - 1) / LSE_BLOCKS)  // 786

typedef float v2f __attribute__((ext_vector_type(2)));
typedef float v8f __attribute__((ext_vector_type(8)));

__device__ __forceinline__ v8f wmma4(v2f a, v2f b, v8f c) {
  // D = A(16x4 f32) * B(4x16 f32) + C(16x16 f32)
  return __builtin_amdgcn_wmma_f32_16x16x4_f32(false, a, false, b, (short)0, c,
                                               false, false);
}

// One wave: y[rowBase..rowBase+15] = W[rows] . xv  (W row-major, stride HDIM).
// B is the vector broadcast across all 16 columns, so every column of D holds y.
// Unroll 8 -> 16 b64 loads in flight, 2 independent WMMA chains.
__device__ __forceinline__ v8f gemv_tile16(const float* __restrict__ W,
                                           long rowBase, long maxRow,
                                           const float* __restrict__ xv) {
  const int lane = threadIdx.x & 31;
  const int half = lane >> 4;   // 0: K=0,1   1: K=2,3 (per 32-bit A layout)
  const int m    = lane & 15;   // row within tile
  long row = rowBase + m;
  if (row > maxRow) row = maxRow;               // clamp (keeps EXEC uniform)
  const float* wr = W + row * (long)HDIM + 2 * half;
  const float* xr = xv + 2 * half;

  v8f acc0 = {0.f, 0.f, 0.f, 0.f, 0.f, 0.f, 0.f, 0.f};
  v8f acc1 = {0.f, 0.f, 0.f, 0.f, 0.f, 0.f, 0.f, 0.f};
#pragma unroll 8
  for (int k = 0; k < HDIM; k += 8) {
    v2f a0 = *(const v2f*)(wr + k);
    v2f b0 = *(const v2f*)(xr + k);
    acc0 = wmma4(a0, b0, acc0);
    v2f a1 = *(const v2f*)(wr + k + 4);
    v2f b1 = *(const v2f*)(xr + k + 4);
    acc1 = wmma4(a1, b1, acc1);
  }
  return acc0 + acc1;
}

// Fused dual-matrix GEMV: y = Wa.xa + Wb.xb for one 16-row tile, single k-loop,
// 4 independent accumulator chains so the scheduler can pipeline both streams.
__device__ __forceinline__ v8f gemv2_tile16(const float* __restrict__ Wa,
                                            const float* __restrict__ xa,
                                            const float* __restrict__ Wb,
                                            const float* __restrict__ xb,
                                            long rowBase, long maxRow) {
  const int lane = threadIdx.x & 31;
  const int half = lane >> 4;
  const int m    = lane & 15;
  long row = rowBase + m;
  if (row > maxRow) row = maxRow;
  const float* wra = Wa + row * (long)HDIM + 2 * half;
  const float* wrb = Wb + row * (long)HDIM + 2 * half;
  const float* xra = xa + 2 * half;
  const float* xrb = xb + 2 * half;

  v8f accA0 = {0.f, 0.f, 0.f, 0.f, 0.f, 0.f, 0.f, 0.f};
  v8f accA1 = {0.f, 0.f, 0.f, 0.f, 0.f, 0.f, 0.f, 0.f};
  v8f accB0 = {0.f, 0.f, 0.f, 0.f, 0.f, 0.f, 0.f, 0.f};
  v8f accB1 = {0.f, 0.f, 0.f, 0.f, 0.f, 0.f, 0.f, 0.f};
#pragma unroll 4
  for (int k = 0; k < HDIM; k += 8) {
    v2f aa0 = *(const v2f*)(wra + k);
    v2f xa0 = *(const v2f*)(xra + k);
    v2f bb0 = *(const v2f*)(wrb + k);
    v2f xb0 = *(const v2f*)(xrb + k);
    accA0 = wmma4(aa0, xa0, accA0);
    accB0 = wmma4(bb0, xb0, accB0);
    v2f aa1 = *(const v2f*)(wra + k + 4);
    v2f xa1 = *(const v2f*)(xra + k + 4);
    v2f bb1 = *(const v2f*)(wrb + k + 4);
    v2f xb1 = *(const v2f*)(xrb + k + 4);
    accA1 = wmma4(aa1, xa1, accA1);
    accB1 = wmma4(bb1, xb1, accB1);
  }
  return (accA0 + accA1) + (accB0 + accB1);
}

// ---- kernel 0: e = relu(emb[x]) ----
__global__ void embed_relu_kernel(const int* __restrict__ x,
                                  const float* __restrict__ emb,
                                  float* __restrict__ evec) {
  int i = blockIdx.x * blockDim.x + threadIdx.x;
  if (i < HDIM) {
    long tok = (long)x[0];
    evec[i] = fmaxf(emb[tok * (long)HDIM + i], 0.0f);
  }
}

// ---- kernel 1: gates = w_ih.e + b_ih + w_hh.h + b_hh  (fused WMMA GEMV) ----
__global__ void gates_kernel(const float* __restrict__ w_ih,
                             const float* __restrict__ w_hh,
                             const float* __restrict__ b_ih,
                             const float* __restrict__ b_hh,
                             const float* __restrict__ evec,
                             const float* __restrict__ hvec,
                             float* __restrict__ gates) {
  long wave = (long)((blockIdx.x * blockDim.x + threadIdx.x) >> 5);
  long rb = wave * 16;
  if (rb >= GDIM) return;                       // wave-uniform exit
  v8f acc = gemv2_tile16(w_ih, evec, w_hh, hvec, rb, GDIM - 1);
  int lane = threadIdx.x & 31;
  if ((lane & 15) == 0) {                       // lanes 0 (M=0..7) and 16 (M=8..15)
    int mbase = (lane >> 4) * 8;
    for (int j = 0; j < 8; ++j) {
      long r = rb + mbase + j;
      gates[r] = acc[j] + b_ih[r] + b_hh[r];
    }
  }
}

// ---- kernel 2: LSTM cell elementwise; writes h_new/c_new into d_out tail ----
__global__ void cell_kernel(const float* __restrict__ gates,
                            const float* __restrict__ c0,
                            float* __restrict__ out,     // d_out base
                            float* __restrict__ hvec) {  // ws copy of h_new
  int t = blockIdx.x * blockDim.x + threadIdx.x;
  if (t >= HDIM) return;
  float gi = gates[t];
  float gf = gates[HDIM + t];
  float gg = gates[2 * HDIM + t];
  float go = gates[3 * HDIM + t];
  float i = 1.0f / (1.0f + expf(-gi));
  float f = 1.0f / (1.0f + expf(-gf));
  float g = tanhf(gg);
  float o = 1.0f / (1.0f + expf(-go));
  float cn = f * c0[t] + i * g;
  float hn = o * tanhf(cn);
  out[VDIM + t] = hn;           // h_new
  out[VDIM + HDIM + t] = cn;    // c_new
  hvec[t] = hn;
}

// ---- kernel 3: logits = w_out.h_new + b_out  (WMMA GEMV, into d_out[0:V)) ----
__global__ void logits_kernel(const float* __restrict__ w_out,
                              const float* __restrict__ b_out,
                              const float* __restrict__ hvec,
                              float* __restrict__ out) {
  long wave = (long)((blockIdx.x * blockDim.x + threadIdx.x) >> 5);
  long rb = wave * 16;
  if (rb >= (long)NTILES_OUT * 16) return;      // wave-uniform exit
  v8f acc = gemv_tile16(w_out, rb, VDIM - 1, hvec);
  int lane = threadIdx.x & 31;
  if ((lane & 15) == 0) {
    int mbase = (lane >> 4) * 8;
    for (int j = 0; j < 8; ++j) {
      long r = rb + mbase + j;
      if (r < VDIM) out[r] = acc[j] + b_out[r];
    }
  }
}

// ---- log-sum-exp (deterministic two-stage streaming reduction) ----
__device__ __forceinline__ void lse_merge(float& m, float& s, float m2, float s2) {
  float M = fmaxf(m, m2);
  if (M == -INFINITY) { m = M; s = 0.0f; return; }
  s = s * expf(m - M) + s2 * expf(m2 - M);
  m = M;
}

__global__ void lse_partial_kernel(const float* __restrict__ logits,
                                   float* __restrict__ partM,
                                   float* __restrict__ partS) {
  __shared__ float sm[256];
  __shared__ float ss[256];
  int b = blockIdx.x, t = threadIdx.x;
  long start = (long)b * LSE_CHUNK;
  long end = start + LSE_CHUNK;
  if (end > VDIM) end = VDIM;
  float m = -INFINITY, s = 0.0f;
  for (long v = start + t; v < end; v += 256) {
    float x = logits[v];
    float M = fmaxf(m, x);
    s = s * expf(m - M) + expf(x - M);
    m = M;
  }
  sm[t] = m; ss[t] = s;
  __syncthreads();
  for (int off = 128; off > 0; off >>= 1) {
    if (t < off) {
      float mm = sm[t], sv = ss[t];
      lse_merge(mm, sv, sm[t + off], ss[t + off]);
      sm[t] = mm; ss[t] = sv;
    }
    __syncthreads();
  }
  if (t == 0) { partM[b] = sm[0]; partS[b] = ss[0]; }
}

__global__ void lse_final_kernel(const float* __restrict__ partM,
                                 const float* __restrict__ partS,
                                 float* __restrict__ lse) {
  __shared__ float sm[LSE_BLOCKS];
  __shared__ float ss[LSE_BLOCKS];
  int t = threadIdx.x;
  sm[t] = partM[t]; ss[t] = partS[t];
  __syncthreads();
  for (int off = LSE_BLOCKS / 2; off > 0; off >>= 1) {
    if (t < off) {
      float mm = sm[t], sv = ss[t];
      lse_merge(mm, sv, sm[t + off], ss[t + off]);
      sm[t] = mm; ss[t] = sv;
    }
    __syncthreads();
  }
  if (t == 0) lse[0] = sm[0] + logf(ss[0]);
}

__global__ void logsoftmax_sub_kernel(float* __restrict__ out,
                                      const float* __restrict__ lse) {
  int v = blockIdx.x * blockDim.x + threadIdx.x;
  if (v < VDIM) out[v] -= lse[0];
}

extern "C" void kernel_launch(void* const* d_in, const int* in_sizes, int n_in,
                              void* d_out, int out_size, void* d_ws, size_t ws_size,
                              hipStream_t stream) {
  const int*   x     = (const int*)  d_in[0];
  const float* h0    = (const float*)d_in[1];
  const float* c0    = (const float*)d_in[2];
  const float* emb   = (const float*)d_in[3];
  const float* w_ih  = (const float*)d_in[4];
  const float* w_hh  = (const float*)d_in[5];
  const float* b_ih  = (const float*)d_in[6];
  const float* b_hh  = (const float*)d_in[7];
  const float* w_out = (const float*)d_in[8];
  const float* b_out = (const float*)d_in[9];
  float* out = (float*)d_out;

  float* ws    = (float*)d_ws;
  float* evec  = ws;            // 1024
  float* hvec  = ws + 1024;     // 1024 (h_new)
  float* gates = ws + 2048;     // 4096
  float* partM = ws + 6144;     // 64
  float* partS = ws + 6208;     // 64
  float* lse   = ws + 6272;     // 1

  embed_relu_kernel<<<4, 256, 0, stream>>>(x, emb, evec);

  // 4096 rows -> 256 wave-tiles, 4 waves (128 thr) per block
  gates_kernel<<<64, 128, 0, stream>>>(w_ih, w_hh, b_ih, b_hh, evec, h0, gates);

  cell_kernel<<<4, 256, 0, stream>>>(gates, c0, out, hvec);

  // 50257 rows -> 3142 wave-tiles -> 786 blocks x 4 waves
  logits_kernel<<<(NTILES_OUT + 3) / 4, 128, 0, stream>>>(w_out, b_out, hvec, out);

  lse_partial_kernel<<<LSE_BLOCKS, 256, 0, stream>>>(out, partM, partS);
  lse_final_kernel<<<1, LSE_BLOCKS, 0, stream>>>(partM, partS, lse);
  logsoftmax_sub_kernel<<<(VDIM + 255) / 256, 256, 0, stream>>>(out, lse);
}